// GraphTransformerEdgeLayer_82497731822090
// MI455X (gfx1250) — compile-verified
//
#include <hip/hip_runtime.h>
#include <math.h>

// ---------------------------------------------------------------------------
// Problem sizes (fixed by the reference)
// ---------------------------------------------------------------------------
#define NNODES 30000
#define NEDGES 480000
#define DIM    128
#define EDIM   64
#define NH     8
#define DHEAD  16

typedef __attribute__((ext_vector_type(2))) float v2f;
typedef __attribute__((ext_vector_type(8))) float v8f;

// ---------------------------------------------------------------------------
// Workspace layout (float offsets), lifetime-based reuse. Total ~538 MB.
// ---------------------------------------------------------------------------
static const size_t O_Q    = 0;                          // 30000*128
static const size_t O_K    = 3840000;                    // 30000*128
static const size_t O_V    = 7680000;                    // 30000*128
static const size_t O_PE   = 11520000;                   // 480000*128 (61.44M)
static const size_t O_SC   = 72960000;                   // 480000*8
static const size_t O_Z    = 76800000;                   // 30000*8
// reuse after attention:
static const size_t O_HATTN = O_Q;
static const size_t O_H1    = O_K;
static const size_t O_HBN   = O_Q;
static const size_t O_HFFN  = O_PE;                      // 30000*256
static const size_t O_H2    = O_K;
// edge path (after node path):
static const size_t O_ETMP  = O_PE;                      // 480000*64
static const size_t O_E1    = 42240000;                  // 480000*64
static const size_t O_EBN   = O_PE;                      // 480000*64
static const size_t O_EFFN  = 42240000;                  // 480000*128 -> ends 103,680,000
static const size_t O_STAT  = 103680000;                 // 2048 floats (4 BN slots)
static const size_t O_E2    = 103682048;                 // 480000*64
static const size_t WS_FLOATS = O_E2 + (size_t)NEDGES * EDIM; // 134,402,048

// ---------------------------------------------------------------------------
// fp32 atomic add lowering to hardware global_atomic_add_f32
// ---------------------------------------------------------------------------
__device__ __forceinline__ void atomAddF(float* p, float v) {
  __hip_atomic_fetch_add(p, v, __ATOMIC_RELAXED, __HIP_MEMORY_SCOPE_AGENT);
}

// ---------------------------------------------------------------------------
// Generic fp32 WMMA GEMM: C[M,N] = A[M,K] @ B[K,N]  (+bias)(+relu)(+residual)
// Optionally accumulates per-channel sum / sum-of-squares for BatchNorm.
// One wave32 computes one 16x16 tile via V_WMMA_F32_16X16X4_F32.
//   A 16x4 layout : lane l (r=l&15, h=l>>4): a.x=A[r,k0+2h], a.y=A[r,k0+2h+1]
//   B 4x16 layout : b.x=B[k0+2h, col],       b.y=B[k0+2h+1, col]
//   C 16x16 layout: acc[i] = C[i + 8h, col]
// ---------------------------------------------------------------------------
template <bool HAS_BIAS, bool RELU, bool HAS_RES, bool STATS>
__global__ __launch_bounds__(256) void gemm_wmma_f32(
    const float* __restrict__ A, const float* __restrict__ B,
    const float* __restrict__ bias, const float* __restrict__ res,
    float* __restrict__ C, float* __restrict__ stat_sum,
    float* __restrict__ stat_sq, int M, int Kd, int Ncols) {
  const int gtid  = blockIdx.x * blockDim.x + threadIdx.x;
  const int wave  = gtid >> 5;
  const int lane  = threadIdx.x & 31;
  const int ntile = Ncols >> 4;
  const int row_t = wave / ntile;
  const int col_t = wave - row_t * ntile;
  if (row_t * 16 >= M) return;   // whole-wave uniform: EXEC stays all-ones

  const int r    = lane & 15;
  const int half = lane >> 4;
  const int col  = (col_t << 4) + r;
  const size_t arow = (size_t)((row_t << 4) + r) * (size_t)Kd;

  v8f acc = {};
  for (int k0 = 0; k0 < Kd; k0 += 4) {
    const int ka = k0 + (half << 1);
    v2f a, b;
    a.x = A[arow + ka];
    a.y = A[arow + ka + 1];
    b.x = B[(size_t)ka * Ncols + col];
    b.y = B[(size_t)(ka + 1) * Ncols + col];
    acc = __builtin_amdgcn_wmma_f32_16x16x4_f32(
        /*neg_a=*/false, a, /*neg_b=*/false, b,
        /*c_mod=*/(short)0, acc, /*reuse_a=*/false, /*reuse_b=*/false);
  }

  const float bval = HAS_BIAS ? bias[col] : 0.0f;
  float psum = 0.0f, psq = 0.0f;
#pragma unroll
  for (int i = 0; i < 8; ++i) {
    const int row = (row_t << 4) + i + (half << 3);
    float v = acc[i] + bval;
    if (RELU) v = fmaxf(v, 0.0f);
    if (HAS_RES) v += res[(size_t)row * Ncols + col];
    C[(size_t)row * Ncols + col] = v;
    if (STATS) { psum += v; psq += v * v; }
  }
  if (STATS) {
    // channel index is the global column; lanes {r, r+16} both contribute
    atomAddF(&stat_sum[col], psum);
    atomAddF(&stat_sq[col], psq);
  }
}

// ---------------------------------------------------------------------------
// score[e,h] = 0.25 * sum_d Q[dst,h,d]*K[src,h,d]*Pe[e,h,d];  z[dst,h] += exp(clip)
// ---------------------------------------------------------------------------
__global__ __launch_bounds__(256) void score_kernel(
    const float* __restrict__ Q, const float* __restrict__ K,
    const float* __restrict__ Pe, const int* __restrict__ src,
    const int* __restrict__ dst, float* __restrict__ score,
    float* __restrict__ z) {
  int idx = blockIdx.x * blockDim.x + threadIdx.x;
  if (idx >= NEDGES * NH) return;
  int e = idx >> 3, h = idx & 7;
  int s = src[e], d = dst[e];
  const float4* q = (const float4*)(Q + (size_t)d * DIM + h * DHEAD);
  const float4* k = (const float4*)(K + (size_t)s * DIM + h * DHEAD);
  const float4* p = (const float4*)(Pe + (size_t)e * DIM + h * DHEAD);
  float acc = 0.0f;
#pragma unroll
  for (int i = 0; i < 4; ++i) {
    float4 a = q[i], b = k[i], c = p[i];
    acc += a.x * b.x * c.x + a.y * b.y * c.y + a.z * b.z * c.z + a.w * b.w * c.w;
  }
  acc *= 0.25f; // 1/sqrt(DHEAD)
  score[idx] = acc;
  float sc = expf(fminf(fmaxf(acc, -5.0f), 5.0f));
  atomAddF(&z[(size_t)d * NH + h], sc);
}

// ---------------------------------------------------------------------------
// h_attn[dst] += V[src] * alpha   (alpha = exp(clip(score)) / (z[dst]+1e-6))
// ---------------------------------------------------------------------------
__global__ __launch_bounds__(256) void aggregate_kernel(
    const float* __restrict__ V, const int* __restrict__ src,
    const int* __restrict__ dst, const float* __restrict__ score,
    const float* __restrict__ z, float* __restrict__ h_attn) {
  int idx = blockIdx.x * blockDim.x + threadIdx.x;
  if (idx >= NEDGES * NH) return;
  int e = idx >> 3, h = idx & 7;
  int s = src[e], d = dst[e];
  float sc = expf(fminf(fmaxf(score[idx], -5.0f), 5.0f));
  float alpha = sc / (z[(size_t)d * NH + h] + 1e-6f);
  const float4* vv = (const float4*)(V + (size_t)s * DIM + h * DHEAD);
  float* out = h_attn + (size_t)d * DIM + h * DHEAD;
#pragma unroll
  for (int i = 0; i < 4; ++i) {
    float4 m = vv[i];
    atomAddF(&out[4 * i + 0], m.x * alpha);
    atomAddF(&out[4 * i + 1], m.y * alpha);
    atomAddF(&out[4 * i + 2], m.z * alpha);
    atomAddF(&out[4 * i + 3], m.w * alpha);
  }
}

// ---------------------------------------------------------------------------
// BatchNorm finalize: scale = g*rsqrt(var+eps), shift = b - mean*scale
// ---------------------------------------------------------------------------
__global__ void bn_finalize_kernel(const float* __restrict__ sum,
                                   const float* __restrict__ sq,
                                   const float* __restrict__ g,
                                   const float* __restrict__ b,
                                   float* __restrict__ scale,
                                   float* __restrict__ shift, int C,
                                   float invM) {
  int c = blockIdx.x * blockDim.x + threadIdx.x;
  if (c >= C) return;
  float m = sum[c] * invM;
  float var = sq[c] * invM - m * m;
  float s = g[c] * rsqrtf(var + 1e-5f);
  scale[c] = s;
  shift[c] = b[c] - m * s;
}

template <int C>
__global__ __launch_bounds__(256) void bn_apply_kernel(
    const float* __restrict__ x, const float* __restrict__ scale,
    const float* __restrict__ shift, float* __restrict__ y, int M) {
  int idx = blockIdx.x * blockDim.x + threadIdx.x;  // one float4 each
  int total = M * (C / 4);
  if (idx >= total) return;
  int c4 = (idx % (C / 4)) * 4;
  float4 v = ((const float4*)x)[idx];
  float4 o;
  o.x = v.x * scale[c4 + 0] + shift[c4 + 0];
  o.y = v.y * scale[c4 + 1] + shift[c4 + 1];
  o.z = v.z * scale[c4 + 2] + shift[c4 + 2];
  o.w = v.w * scale[c4 + 3] + shift[c4 + 3];
  ((float4*)y)[idx] = o;
}

// ---------------------------------------------------------------------------
// Host-side orchestration
// ---------------------------------------------------------------------------
extern "C" void kernel_launch(void* const* d_in, const int* in_sizes, int n_in,
                              void* d_out, int out_size, void* d_ws,
                              size_t ws_size, hipStream_t stream) {
  (void)in_sizes; (void)n_in; (void)out_size;
  if (ws_size < WS_FLOATS * sizeof(float)) return;  // workspace too small

  const float* x    = (const float*)d_in[0];
  const int*   ei   = (const int*)d_in[1];
  const float* ea   = (const float*)d_in[2];
  const float* Wq   = (const float*)d_in[3];
  const float* Wk   = (const float*)d_in[4];
  const float* Wv   = (const float*)d_in[5];
  const float* We   = (const float*)d_in[6];
  const float* W_Oh = (const float*)d_in[7];
  const float* b_Oh = (const float*)d_in[8];
  const float* W_ep = (const float*)d_in[9];
  const float* b_ep = (const float*)d_in[10];
  const float* W_Oe = (const float*)d_in[11];
  const float* b_Oe = (const float*)d_in[12];
  const float* W_h1 = (const float*)d_in[13];
  const float* b_h1 = (const float*)d_in[14];
  const float* W_h2 = (const float*)d_in[15];
  const float* b_h2 = (const float*)d_in[16];
  const float* W_e1 = (const float*)d_in[17];
  const float* b_e1 = (const float*)d_in[18];
  const float* W_e2 = (const float*)d_in[19];
  const float* b_e2 = (const float*)d_in[20];
  const float* g1h  = (const float*)d_in[21];
  const float* B1h  = (const float*)d_in[22];
  const float* g1e  = (const float*)d_in[23];
  const float* B1e  = (const float*)d_in[24];
  const float* g2h  = (const float*)d_in[25];
  const float* B2h  = (const float*)d_in[26];
  const float* g2e  = (const float*)d_in[27];
  const float* B2e  = (const float*)d_in[28];

  float* ws    = (float*)d_ws;
  float* out_h = (float*)d_out;
  float* out_e = out_h + (size_t)NNODES * DIM;

  const int* src = ei;
  const int* dst = ei + NEDGES;

  float* Q     = ws + O_Q;
  float* Kbuf  = ws + O_K;
  float* V     = ws + O_V;
  float* Pe    = ws + O_PE;
  float* score = ws + O_SC;
  float* z     = ws + O_Z;
  float* ST    = ws + O_STAT;
  // BN stat slots: [sum(128) | sq(128) | scale(128) | shift(128)] x 4
  float* sum0 = ST + 0,    *sq0 = ST + 128,  *scl0 = ST + 256,  *shf0 = ST + 384;
  float* sum1 = ST + 512,  *sq1 = ST + 640,  *scl1 = ST + 768,  *shf1 = ST + 896;
  float* sum2 = ST + 1024, *sq2 = ST + 1152, *scl2 = ST + 1280, *shf2 = ST + 1408;
  float* sum3 = ST + 1536, *sq3 = ST + 1664, *scl3 = ST + 1792, *shf3 = ST + 1920;

  float* h_attn = ws + O_HATTN;
  float* h1     = ws + O_H1;
  float* hbn    = ws + O_HBN;
  float* hffn   = ws + O_HFFN;
  float* h2     = ws + O_H2;
  float* etmp   = ws + O_ETMP;
  float* e1     = ws + O_E1;
  float* ebn    = ws + O_EBN;
  float* effn   = ws + O_EFFN;
  float* e2     = ws + O_E2;

  auto gblocks = [](int M, int Nc) {
    long waves = (long)(M >> 4) * (Nc >> 4);
    return (int)((waves * 32 + 255) / 256);
  };
  const int EH_BLOCKS = (NEDGES * NH + 255) / 256;

  // zero accumulators (stream-ordered, graph-capture safe)
  hipMemsetAsync(ST, 0, 2048 * sizeof(float), stream);
  hipMemsetAsync(z, 0, (size_t)NNODES * NH * sizeof(float), stream);

  // --- projections -------------------------------------------------------
  gemm_wmma_f32<false, false, false, false><<<gblocks(NNODES, DIM), 256, 0, stream>>>(
      x, Wq, nullptr, nullptr, Q, nullptr, nullptr, NNODES, DIM, DIM);
  gemm_wmma_f32<false, false, false, false><<<gblocks(NNODES, DIM), 256, 0, stream>>>(
      x, Wk, nullptr, nullptr, Kbuf, nullptr, nullptr, NNODES, DIM, DIM);
  gemm_wmma_f32<false, false, false, false><<<gblocks(NNODES, DIM), 256, 0, stream>>>(
      x, Wv, nullptr, nullptr, V, nullptr, nullptr, NNODES, DIM, DIM);
  gemm_wmma_f32<false, false, false, false><<<gblocks(NEDGES, DIM), 256, 0, stream>>>(
      ea, We, nullptr, nullptr, Pe, nullptr, nullptr, NEDGES, EDIM, DIM);

  // --- attention ---------------------------------------------------------
  score_kernel<<<EH_BLOCKS, 256, 0, stream>>>(Q, Kbuf, Pe, src, dst, score, z);
  hipMemsetAsync(h_attn, 0, (size_t)NNODES * DIM * sizeof(float), stream); // reuses Q region
  aggregate_kernel<<<EH_BLOCKS, 256, 0, stream>>>(V, src, dst, score, z, h_attn);

  // --- node path: O-proj + residual + BN1 --------------------------------
  gemm_wmma_f32<true, false, true, true><<<gblocks(NNODES, DIM), 256, 0, stream>>>(
      h_attn, W_Oh, b_Oh, x, h1, sum0, sq0, NNODES, DIM, DIM);
  bn_finalize_kernel<<<1, 128, 0, stream>>>(sum0, sq0, g1h, B1h, scl0, shf0, DIM,
                                            1.0f / NNODES);
  bn_apply_kernel<DIM><<<(NNODES * DIM / 4 + 255) / 256, 256, 0, stream>>>(
      h1, scl0, shf0, hbn, NNODES);

  // --- node FFN + residual + BN2 -> out_h --------------------------------
  gemm_wmma_f32<true, true, false, false><<<gblocks(NNODES, 2 * DIM), 256, 0, stream>>>(
      hbn, W_h1, b_h1, nullptr, hffn, nullptr, nullptr, NNODES, DIM, 2 * DIM);
  gemm_wmma_f32<true, false, true, true><<<gblocks(NNODES, DIM), 256, 0, stream>>>(
      hffn, W_h2, b_h2, hbn, h2, sum1, sq1, NNODES, 2 * DIM, DIM);
  bn_finalize_kernel<<<1, 128, 0, stream>>>(sum1, sq1, g2h, B2h, scl1, shf1, DIM,
                                            1.0f / NNODES);
  bn_apply_kernel<DIM><<<(NNODES * DIM / 4 + 255) / 256, 256, 0, stream>>>(
      h2, scl1, shf1, out_h, NNODES);

  // --- edge path: score proj -> O-proj + residual + BN1 ------------------
  gemm_wmma_f32<true, false, false, false><<<gblocks(NEDGES, EDIM), 256, 0, stream>>>(
      score, W_ep, b_ep, nullptr, etmp, nullptr, nullptr, NEDGES, NH, EDIM);
  gemm_wmma_f32<true, false, true, true><<<gblocks(NEDGES, EDIM), 256, 0, stream>>>(
      etmp, W_Oe, b_Oe, ea, e1, sum2, sq2, NEDGES, EDIM, EDIM);
  bn_finalize_kernel<<<1, 64, 0, stream>>>(sum2, sq2, g1e, B1e, scl2, shf2, EDIM,
                                           1.0f / NEDGES);
  bn_apply_kernel<EDIM><<<(NEDGES * EDIM / 4 + 255) / 256, 256, 0, stream>>>(
      e1, scl2, shf2, ebn, NEDGES);

  // --- edge FFN + residual + BN2 -> out_e --------------------------------
  gemm_wmma_f32<true, true, false, false><<<gblocks(NEDGES, 2 * EDIM), 256, 0, stream>>>(
      ebn, W_e1, b_e1, nullptr, effn, nullptr, nullptr, NEDGES, EDIM, 2 * EDIM);
  gemm_wmma_f32<true, false, true, true><<<gblocks(NEDGES, EDIM), 256, 0, stream>>>(
      effn, W_e2, b_e2, ebn, e2, sum3, sq3, NEDGES, 2 * EDIM, EDIM);
  bn_finalize_kernel<<<1, 64, 0, stream>>>(sum3, sq3, g2e, B2e, scl3, shf3, EDIM,
                                           1.0f / NEDGES);
  bn_apply_kernel<EDIM><<<(NEDGES * EDIM / 4 + 255) / 256, 256, 0, stream>>>(
      e2, scl3, shf3, out_e, NEDGES);
}